// Model_3470333575382
// MI455X (gfx1250) — compile-verified
//
#include <hip/hip_runtime.h>

// Problem constants (from reference): B=64, H=32, D=128, MAX_SEQ=512
#define B_ 64
#define H_ 32
#define D_ 128

// LDS row stride in dwords: 128 data + 16 pad (TDM pad_enable) -> conflict-free ds reads
#define ROWSTRIDE 144
#define CHUNK_ROWS 16
#define NCHUNK (D_ / CHUNK_ROWS)

typedef __attribute__((ext_vector_type(2))) float v2f;
typedef __attribute__((ext_vector_type(8))) float v8f;
typedef __attribute__((ext_vector_type(4))) unsigned int u32x4;
typedef __attribute__((ext_vector_type(8))) int i32x8;
typedef __attribute__((ext_vector_type(4))) int i32x4;

// Issue one TDM load: 16 rows x 128 f32 (8 KB) from global -> LDS with
// 16-dword padding inserted after every 128 dwords (row stride 144 dw).
__device__ __forceinline__ void tdm_load_tile(const float* gsrc, unsigned lds_off)
{
    const unsigned long long ga = (unsigned long long)(uintptr_t)gsrc;

    // D# group 0: count=1 | lds_addr | global_addr[56:0] | type=2 ("image")
    u32x4 g0;
    g0[0] = 1u;                                                  // count=1, user desc
    g0[1] = lds_off;                                             // lds_addr (bytes)
    g0[2] = (unsigned)ga;                                        // global_addr[31:0]
    g0[3] = (unsigned)((ga >> 32) & 0x01FFFFFFu) | (2u << 30);   // addr[56:32], type=2

    // D# group 1: data_size=4B (2), pad_enable, pad_interval=128dw (6),
    // pad_amount=16dw (15); tensor 128 x 16, tile 128 x 16, dim0_stride=128.
    i32x8 g1;
    g1[0] = (int)((2u << 16) | (1u << 20) | (6u << 22) | (15u << 25));
    g1[1] = (int)(128u << 16);   // tensor_dim0 = 128      (bits 79:48, low half)
    g1[2] = (int)(16u  << 16);   // tensor_dim1 = 16       (bits 111:80, low half)
    g1[3] = (int)(128u << 16);   // tile_dim0   = 128      (bits 127:112)
    g1[4] = (int)16;             // tile_dim1   = 16, tile_dim2 = 0
    g1[5] = (int)128;            // tensor_dim0_stride = 128 (bits 207:160, low)
    g1[6] = 0;
    g1[7] = 0;                   // tensor_dim1_stride = 0 (2D)

    const i32x4 gz4 = {0, 0, 0, 0};
    const i32x8 gz8 = {0, 0, 0, 0, 0, 0, 0, 0};
    __builtin_amdgcn_tensor_load_to_lds(g0, g1, gz4, gz4, gz8, 0);
}

// One workgroup (128 threads = 4 wave32) per (b,h).
// out[b,h,e] = valid ? decay_h * sum_d q[d]*state[d,e] + (q.k)*v[e] : 0
__global__ __launch_bounds__(128) void linattn_decode_wmma(
    const float* __restrict__ q,
    const float* __restrict__ k,
    const float* __restrict__ v,
    const float* __restrict__ kv_cache,
    const float* __restrict__ slope_rate,
    const long long* __restrict__ slot_idx,
    float* __restrict__ out)
{
    __shared__ float smem[2][CHUNK_ROWS * ROWSTRIDE];  // double-buffered state chunks
    __shared__ float qs[D_];
    __shared__ float red[128];
    __shared__ float qk_sh;

    const int bh  = blockIdx.x;
    const int b   = bh / H_;
    const int h   = bh % H_;
    const int tid = threadIdx.x;

    const long long slot  = slot_idx[b];
    const bool      valid = (slot >= 0);
    const long long safe  = valid ? slot : 0;

    const float* qrow = q + (size_t)bh * D_;
    const float* krow = k + (size_t)bh * D_;
    const float* vrow = v + (size_t)bh * D_;
    const float* st   = kv_cache + ((size_t)safe * H_ + h) * (size_t)(D_ * D_);

    const unsigned lds0 = (unsigned)(uintptr_t)&smem[0][0];
    const unsigned lds1 = (unsigned)(uintptr_t)&smem[1][0];

    // Kick off DMA of chunk 0 (wave 0 only; TDM ignores EXEC, gate by wave).
    if (tid < 32) {
        tdm_load_tile(st, lds0);
    }

    // Stage q into LDS; compute qk = dot(q,k) with a wave reduction.
    const float qv = qrow[tid];
    qs[tid]  = qv;
    red[tid] = qv * krow[tid];
    __syncthreads();
    if (tid < 32) {
        float s = red[tid] + red[tid + 32] + red[tid + 64] + red[tid + 96];
        #pragma unroll
        for (int off = 16; off > 0; off >>= 1) s += __shfl_down(s, off, 32);
        if (tid == 0) qk_sh = s;
    }
    __syncthreads();

    const float qk    = qk_sh;
    const float decay = __expf(-slope_rate[h]);

    const int wave = tid >> 5;
    const int lane = tid & 31;
    const int n    = lane & 15;   // N column within 16-wide tile
    const int kh   = lane >> 4;   // K half: lanes 0-15 -> K{0,1}, lanes 16-31 -> K{2,3}

    // Each wave produces 32 output columns: [e0, e0+32)
    const int e0 = wave * 32;

    v8f c0 = {};  // columns e0 .. e0+15
    v8f c1 = {};  // columns e0+16 .. e0+31

    for (int ch = 0; ch < NCHUNK; ++ch) {
        // Wave 0: prefetch next chunk into the other buffer, then wait until
        // the current chunk's DMA (older, in-order on TENSORcnt) completed.
        if (tid < 32) {
            if (ch + 1 < NCHUNK) {
                tdm_load_tile(st + (size_t)(ch + 1) * CHUNK_ROWS * D_,
                              ((ch + 1) & 1) ? lds1 : lds0);
                __builtin_amdgcn_s_wait_tensorcnt(1);
            } else {
                __builtin_amdgcn_s_wait_tensorcnt(0);
            }
        }
        __syncthreads();  // chunk ch visible to all waves

        const float* buf = &smem[ch & 1][0];
        const float* qsc = qs + ch * CHUNK_ROWS;

        #pragma unroll
        for (int ks = 0; ks < CHUNK_ROWS / 4; ++ks) {
            const int r0 = ks * 4 + 2 * kh;   // this half-wave's first K row in chunk

            // A = q chunk replicated over all 16 M rows (16x4 f32 A layout)
            v2f a;
            a.x = qsc[r0];
            a.y = qsc[r0 + 1];

            // B = 4x16 state chunk from padded LDS (conflict-free banks)
            v2f b0v, b1v;
            b0v.x = buf[(r0)     * ROWSTRIDE + e0 + n];
            b0v.y = buf[(r0 + 1) * ROWSTRIDE + e0 + n];
            b1v.x = buf[(r0)     * ROWSTRIDE + e0 + n + 16];
            b1v.y = buf[(r0 + 1) * ROWSTRIDE + e0 + n + 16];

            c0 = __builtin_amdgcn_wmma_f32_16x16x4_f32(false, a, false, b0v,
                                                       (short)0, c0, false, false);
            c1 = __builtin_amdgcn_wmma_f32_16x16x4_f32(false, a, false, b1v,
                                                       (short)0, c1, false, false);
        }
        __syncthreads();  // all waves done reading buf before it is re-filled
    }

    // All M rows of C are identical; VGPR0 lanes 0-15 hold (M=0, N=lane).
    if (lane < 16) {
        const size_t ob = (size_t)bh * D_ + e0 + n;
        out[ob]      = valid ? fmaf(qk, vrow[e0 + n],      decay * c0[0]) : 0.0f;
        out[ob + 16] = valid ? fmaf(qk, vrow[e0 + 16 + n], decay * c1[0]) : 0.0f;
    }
}

extern "C" void kernel_launch(void* const* d_in, const int* in_sizes, int n_in,
                              void* d_out, int out_size, void* d_ws, size_t ws_size,
                              hipStream_t stream) {
    (void)in_sizes; (void)n_in; (void)out_size; (void)d_ws; (void)ws_size;

    const float*     q        = (const float*)d_in[0];
    const float*     k        = (const float*)d_in[1];
    const float*     v        = (const float*)d_in[2];
    const float*     kv_cache = (const float*)d_in[3];
    const float*     slope    = (const float*)d_in[4];
    const long long* slot     = (const long long*)d_in[5];
    float*           out      = (float*)d_out;

    dim3 grid(B_ * H_);   // 2048 workgroups, one per (b,h)
    dim3 block(128);      // 4 wave32
    hipLaunchKernelGGL(linattn_decode_wmma, grid, block, 0, stream,
                       q, k, v, kv_cache, slope, slot, out);
}